// FactDiagram_87531433492874
// MI455X (gfx1250) — compile-verified
//
#include <hip/hip_runtime.h>
#include <hip/hip_bf16.h>
#include <math.h>

// ---------------- problem constants (match reference) ----------------
#define B_DIM      128
#define INPUT_DIM  256
#define N_THRESH   6
#define NUM_FACTS  1536          // INPUT_DIM * N_THRESH
#define FIN        3072          // 2 * NUM_FACTS
#define N_NODES    1023
#define NPAD       1024          // padded node dim (col 1023 = zero)
#define N_LEAVES   1024
#define OUT_DIM    8
#define KCAT       2048          // [log_t | log_1mt] padded
#define TAU_INV    (1.0f / 0.7f)
#define EPS_T      1e-6f

typedef float v2f __attribute__((ext_vector_type(2)));
typedef float v8f __attribute__((ext_vector_type(8)));

// ---------------- workspace layout (floats) ----------------
// selT   : [FIN][NPAD]      = 3,145,728 f  (12 MB)
// f_aug  : [B][FIN]         =   393,216 f  (1.5 MB)
// logcat : [B][KCAT]        =   262,144 f  (1 MB)
// wcatT  : [KCAT][N_LEAVES] = 2,097,152 f  (8 MB)
// leaf_p : [B][N_LEAVES]    =   131,072 f  (0.5 MB)
#define WS_SELT   0
#define WS_FAUG   (WS_SELT + FIN * NPAD)
#define WS_LOGCAT (WS_FAUG + B_DIM * FIN)
#define WS_WCATT  (WS_LOGCAT + B_DIM * KCAT)
#define WS_LEAFP  (WS_WCATT + KCAT * N_LEAVES)

// ---------------- kernel 1: threshold fact bank ----------------
__global__ void fd_facts(const float* __restrict__ x,
                         const float* __restrict__ th,
                         const float* __restrict__ sl,
                         float* __restrict__ f_aug) {
    int idx = blockIdx.x * blockDim.x + threadIdx.x;   // B*NUM_FACTS threads
    if (idx >= B_DIM * NUM_FACTS) return;
    int b = idx / NUM_FACTS;
    int j = idx % NUM_FACTS;          // j = d*N_THRESH + t  (matches reshape)
    int d = j / N_THRESH;
    int t = j % N_THRESH;
    float z = (x[b * INPUT_DIM + d] - th[d * N_THRESH + t]) * sl[d * N_THRESH + t];
    float f = 1.0f / (1.0f + expf(-z));
    f_aug[b * FIN + j]             = f;
    f_aug[b * FIN + NUM_FACTS + j] = 1.0f - f;
}

// ---------------- kernel 2: softmax over FIN, transposed into selT ---------
// one 256-thread block per padded node row; row N_NODES..NPAD-1 writes zeros
__global__ void fd_softmax(const float* __restrict__ sel_logits,
                           float* __restrict__ selT) {
    __shared__ float red[256];
    int n   = blockIdx.x;
    int tid = threadIdx.x;
    if (n >= N_NODES) {                        // zero pad column
        for (int k = tid; k < FIN; k += 256) selT[k * NPAD + n] = 0.0f;
        return;
    }
    const float* l = sel_logits + (size_t)n * FIN;
    float v = -INFINITY;
    for (int k = tid; k < FIN; k += 256) v = fmaxf(v, l[k]);
    red[tid] = v; __syncthreads();
    for (int s = 128; s > 0; s >>= 1) {
        if (tid < s) red[tid] = fmaxf(red[tid], red[tid + s]);
        __syncthreads();
    }
    float lmax = red[0]; __syncthreads();
    float acc = 0.0f;
    for (int k = tid; k < FIN; k += 256) acc += expf((l[k] - lmax) * TAU_INV);
    red[tid] = acc; __syncthreads();
    for (int s = 128; s > 0; s >>= 1) {
        if (tid < s) red[tid] += red[tid + s];
        __syncthreads();
    }
    float inv_sum = 1.0f / red[0];
    for (int k = tid; k < FIN; k += 256)
        selT[k * NPAD + n] = expf((l[k] - lmax) * TAU_INV) * inv_sum;
}

// ---------------- kernel 3: build [W1|W0]^T with node padding ---------------
__global__ void fd_wcat(const float* __restrict__ leaf_used,
                        const float* __restrict__ leaf_dir,
                        float* __restrict__ wcatT) {
    int idx = blockIdx.x * blockDim.x + threadIdx.x;   // KCAT*N_LEAVES threads
    if (idx >= KCAT * N_LEAVES) return;
    int k    = idx / N_LEAVES;
    int leaf = idx % N_LEAVES;
    int node = k & (NPAD - 1);
    int half = k >> 10;                                // 0: W1, 1: W0
    float v = 0.0f;
    if (node < N_NODES) {
        float u = leaf_used[leaf * N_NODES + node];
        float d = leaf_dir [leaf * N_NODES + node];
        v = half == 0 ? u * d : u * (1.0f - d);
    }
    wcatT[k * N_LEAVES + leaf] = v;
}

// ---------------- fp32 WMMA tile engine -------------------------------------
// One wave (32 lanes, EXEC all ones) computes a 16(M) x 64(N) f32 tile.
// Fragment layouts per CDNA5 ISA §7.12.2:
//   A 16x4 : lane%16 = M ; VGPR i holds K = i + 2*(lane>=16)
//   B 4x16 : lane%16 = N ; VGPR i holds K = i + 2*(lane>=16)  (symmetric)
//   C 16x16: lane%16 = N ; VGPR i holds M = i + 8*(lane>=16)
__device__ __forceinline__ void fd_wmma_tile_f32(
    const float* __restrict__ A, int lda,   // [M x K] row-major
    const float* __restrict__ Bm, int ldb,  // [K x N] row-major
    int m0, int n0, int K, v8f acc[4]) {
    int lane  = threadIdx.x & 31;
    int l16   = lane & 15;
    int kbase = (lane >> 4) << 1;           // 0 for lanes 0-15, 2 for 16-31
    const float* arow = A + (size_t)(m0 + l16) * lda + kbase;
    for (int k0 = 0; k0 < K; k0 += 4) {
        v2f a;
        a.x = arow[k0];
        a.y = arow[k0 + 1];
        #pragma unroll
        for (int j = 0; j < 4; ++j) {
            int n = n0 + j * 16 + l16;
            v2f b;
            b.x = Bm[(size_t)(k0 + kbase)     * ldb + n];
            b.y = Bm[(size_t)(k0 + kbase + 1) * ldb + n];
            acc[j] = __builtin_amdgcn_wmma_f32_16x16x4_f32(
                false, a, false, b, (short)0, acc[j], false, false);
        }
    }
}

// ---------------- kernel 4: GEMM1 + clip/log epilogue -----------------------
// t = f_aug @ selT ; logcat[b][n] = log t ; logcat[b][NPAD+n] = log1p(-t)
__global__ void __launch_bounds__(32)
fd_gemm1(const float* __restrict__ f_aug,
         const float* __restrict__ selT,
         float* __restrict__ logcat) {
    v8f acc[4] = {};
    int m0 = blockIdx.y * 16;
    int n0 = blockIdx.x * 64;
    fd_wmma_tile_f32(f_aug, FIN, selT, NPAD, m0, n0, FIN, acc);
    int lane = threadIdx.x & 31;
    int l16  = lane & 15;
    int row0 = m0 + ((lane >> 4) << 3);
    #pragma unroll
    for (int j = 0; j < 4; ++j) {
        int n = n0 + j * 16 + l16;
        #pragma unroll
        for (int i = 0; i < 8; ++i) {
            float t = acc[j][i];
            t = fminf(fmaxf(t, EPS_T), 1.0f - EPS_T);
            int r = row0 + i;
            logcat[(size_t)r * KCAT + n]        = logf(t);
            logcat[(size_t)r * KCAT + NPAD + n] = log1pf(-t);
        }
    }
}

// ---------------- kernel 5: GEMM2 + exp epilogue ----------------------------
// leaf_p = exp(logcat @ wcatT)
__global__ void __launch_bounds__(32)
fd_gemm2(const float* __restrict__ logcat,
         const float* __restrict__ wcatT,
         float* __restrict__ leaf_p) {
    v8f acc[4] = {};
    int m0 = blockIdx.y * 16;
    int n0 = blockIdx.x * 64;
    fd_wmma_tile_f32(logcat, KCAT, wcatT, N_LEAVES, m0, n0, KCAT, acc);
    int lane = threadIdx.x & 31;
    int l16  = lane & 15;
    int row0 = m0 + ((lane >> 4) << 3);
    #pragma unroll
    for (int j = 0; j < 4; ++j) {
        int n = n0 + j * 16 + l16;
        #pragma unroll
        for (int i = 0; i < 8; ++i)
            leaf_p[(size_t)(row0 + i) * N_LEAVES + n] = expf(acc[j][i]);
    }
}

// ---------------- kernel 6: final tiny GEMM ---------------------------------
__global__ void fd_out(const float* __restrict__ leaf_p,
                       const float* __restrict__ leaf_value,
                       float* __restrict__ out) {
    int idx = blockIdx.x * blockDim.x + threadIdx.x;   // B*OUT_DIM threads
    if (idx >= B_DIM * OUT_DIM) return;
    int b = idx / OUT_DIM;
    int o = idx % OUT_DIM;
    float s = 0.0f;
    const float* lp = leaf_p + (size_t)b * N_LEAVES;
    for (int l = 0; l < N_LEAVES; ++l)
        s += lp[l] * leaf_value[l * OUT_DIM + o];
    out[b * OUT_DIM + o] = s;
}

// ---------------- launcher --------------------------------------------------
extern "C" void kernel_launch(void* const* d_in, const int* in_sizes, int n_in,
                              void* d_out, int out_size, void* d_ws, size_t ws_size,
                              hipStream_t stream) {
    const float* x          = (const float*)d_in[0];
    const float* thresholds = (const float*)d_in[1];
    const float* slopes     = (const float*)d_in[2];
    const float* sel_logits = (const float*)d_in[3];
    const float* leaf_value = (const float*)d_in[4];
    const float* leaf_used  = (const float*)d_in[5];
    const float* leaf_dir   = (const float*)d_in[6];
    float* out = (float*)d_out;

    float* ws     = (float*)d_ws;
    float* selT   = ws + WS_SELT;
    float* f_aug  = ws + WS_FAUG;
    float* logcat = ws + WS_LOGCAT;
    float* wcatT  = ws + WS_WCATT;
    float* leaf_p = ws + WS_LEAFP;

    // facts: B*NUM_FACTS threads
    fd_facts<<<(B_DIM * NUM_FACTS + 255) / 256, 256, 0, stream>>>(
        x, thresholds, slopes, f_aug);

    // softmax: one block per padded node row
    fd_softmax<<<NPAD, 256, 0, stream>>>(sel_logits, selT);

    // wcat: KCAT*N_LEAVES threads
    fd_wcat<<<(KCAT * N_LEAVES + 255) / 256, 256, 0, stream>>>(
        leaf_used, leaf_dir, wcatT);

    // GEMM1: [128,3072] @ [3072,1024]; grid = (1024/64, 128/16)
    fd_gemm1<<<dim3(NPAD / 64, B_DIM / 16), 32, 0, stream>>>(f_aug, selT, logcat);

    // GEMM2: [128,2048] @ [2048,1024]
    fd_gemm2<<<dim3(N_LEAVES / 64, B_DIM / 16), 32, 0, stream>>>(logcat, wcatT, leaf_p);

    // output: [128,1024] @ [1024,8]
    fd_out<<<(B_DIM * OUT_DIM + 255) / 256, 256, 0, stream>>>(
        leaf_p, leaf_value, out);
}